// ParallelProj2D_29850022707683
// MI455X (gfx1250) — compile-verified
//
#include <hip/hip_runtime.h>
#include <stdint.h>

// ParallelProj2D: ray-driven parallel-beam forward projection, 2x[256,256] -> 2x[180,363].
// TDM-DMA the whole 256x256 f32 image into LDS with hardware row padding
// (256->257 dwords) so gathers are bank-conflict-free at every angle; each
// thread integrates one ray with bilinear gathers from LDS.

#define VOLN        256
#define NANG        180
#define NDET        363
#define NSAMP       363
#define LDS_PITCH   257                       // padded row pitch in floats
#define LDS_BYTES   (VOLN * LDS_PITCH * 4)    // 263168 B (<320KB WGP LDS)
#define RAYS        (NANG * NDET)             // 65340 per batch
#define BLOCK_T     1024                      // 32 wave32s; 1 WG/WGP (LDS-limited)

typedef unsigned int u32;
typedef __attribute__((ext_vector_type(4))) u32 v4u;
typedef __attribute__((ext_vector_type(8))) int v8i;
typedef __attribute__((ext_vector_type(4))) int v4i;

__global__ __launch_bounds__(BLOCK_T)
void ParallelProj2D_kernel(const float* __restrict__ img, float* __restrict__ out) {
    extern __shared__ float s_img[];   // [256][257] via TDM pad
    const int b = blockIdx.y;
    const float* src = img + (size_t)b * VOLN * VOLN;

    // ---- Stage image into LDS with the Tensor Data Mover (wave 0 issues) ----
    if (threadIdx.x < 32) {
        uint64_t ga = (uint64_t)(uintptr_t)src;
        u32 lds_off = (u32)(uintptr_t)(void*)&s_img[0];   // low 32b of flat LDS addr = LDS offset

        v4u g0;
        g0.x = 1u;                                        // count=1 (valid D#), user mode
        g0.y = lds_off;                                   // lds_addr [63:32]
        g0.z = (u32)ga;                                   // global_addr lo
        g0.w = ((u32)(ga >> 32) & 0x01FFFFFFu) | (2u << 30); // global_addr hi | type=2

        v8i g1;
        g1[0] = (int)0x01D20000u;  // wg_mask=0 | data_size=2(4B) | pad_enable | pad_interval=7(256dw) | pad_amount=0(1dw)
        g1[1] = (int)(VOLN << 16); // tensor_dim0 lo16 @ bits[63:48] = 256
        g1[2] = (int)(VOLN << 16); // tensor_dim1 lo16 @ bits[95:80] = 256
        g1[3] = (int)(VOLN << 16); // tile_dim0 @ bits[127:112] = 256
        g1[4] = VOLN;              // tile_dim1 @ bits[143:128] = 256, tile_dim2 = 0
        g1[5] = VOLN;              // tensor_dim0_stride lo32 = 256 elements
        g1[6] = 0;                 // stride hi, tensor_dim1_stride lo
        g1[7] = 0;

        v4i gz = {0, 0, 0, 0};     // 2D tensor: groups 2/3 unused
#if defined(__clang_major__) && (__clang_major__ >= 23)
        v8i z8 = {0, 0, 0, 0, 0, 0, 0, 0};
        __builtin_amdgcn_tensor_load_to_lds(g0, g1, gz, gz, z8, 0);
#else
        __builtin_amdgcn_tensor_load_to_lds(g0, g1, gz, gz, 0);
#endif
        __builtin_amdgcn_s_wait_tensorcnt(0);
    }
    __syncthreads();

    // ---- One ray per thread ----
    const int r = blockIdx.x * BLOCK_T + (int)threadIdx.x;
    if (r >= RAYS) return;
    const int a = r / NDET;
    const int d = r - a * NDET;

    const float theta = (float)a * 0.017453292519943295f;  // pi/180, endpoint=False
    const float c  = cosf(theta);
    const float si = sinf(theta);
    const float s  = (float)d - 181.0f;                    // (NDET-1)/2
    const float sc = s * c;
    const float ss = s * si;

    float acc = 0.0f;
    float t = -181.0f;                                     // (NSAMP-1)/2; exact fp32 counter
#pragma unroll 4
    for (int j = 0; j < NSAMP; ++j, t += 1.0f) {
        const float px = fmaf(-t, si, sc);
        const float py = fmaf( t, c,  ss);
        const float fx = px + 127.5f;                      // (VOL-1)/2
        const float fy = py + 127.5f;
        const float x0f = floorf(fx), y0f = floorf(fy);
        const float wx = fx - x0f,    wy = fy - y0f;
        const int x0 = (int)x0f, y0 = (int)y0f;

        // Validity via unsigned compares (negative -> huge -> invalid).
        const bool vx0 = (unsigned)x0       < (unsigned)VOLN;
        const bool vx1 = (unsigned)(x0 + 1) < (unsigned)VOLN;
        const bool vy0 = (unsigned)y0       < (unsigned)VOLN;
        const bool vy1 = (unsigned)(y0 + 1) < (unsigned)VOLN;

        // Single base address; neighbors become ds_load immediate offsets.
        // Invalid lanes read pad/OOB (LDS OOB loads return 0) and are cndmask'd away.
        const int base = y0 * LDS_PITCH + x0;
        float v00 = s_img[base];
        float v10 = s_img[base + 1];
        float v01 = s_img[base + LDS_PITCH];
        float v11 = s_img[base + LDS_PITCH + 1];
        v00 = (vx0 & vy0) ? v00 : 0.0f;
        v10 = (vx1 & vy0) ? v10 : 0.0f;
        v01 = (vx0 & vy1) ? v01 : 0.0f;
        v11 = (vx1 & vy1) ? v11 : 0.0f;

        const float top = fmaf(wx, v10 - v00, v00);
        const float bot = fmaf(wx, v11 - v01, v01);
        acc = fmaf(wy, bot - top, acc + top);
    }
    out[(size_t)b * RAYS + r] = acc;  // * STEP (=1.0)
}

extern "C" void kernel_launch(void* const* d_in, const int* in_sizes, int n_in,
                              void* d_out, int out_size, void* d_ws, size_t ws_size,
                              hipStream_t stream) {
    (void)in_sizes; (void)n_in; (void)out_size; (void)d_ws; (void)ws_size;
    const float* img = (const float*)d_in[0];
    float* out = (float*)d_out;

    // Dynamic LDS > 64KB requires opting in; deterministic + capture-safe (not a stream op).
    (void)hipFuncSetAttribute((const void*)ParallelProj2D_kernel,
                              hipFuncAttributeMaxDynamicSharedMemorySize, LDS_BYTES);

    dim3 grid((RAYS + BLOCK_T - 1) / BLOCK_T, 2);
    ParallelProj2D_kernel<<<grid, BLOCK_T, LDS_BYTES, stream>>>(img, out);
}